// CrfDecoderScanABC_27118423507517
// MI455X (gfx1250) — compile-verified
//
#include <hip/hip_runtime.h>

typedef float v2f __attribute__((ext_vector_type(2)));
typedef float v8f __attribute__((ext_vector_type(8)));

#define Bn 256
#define Tn 512
#define Cn 4
#define Nn 64
#define PSTR 66  // padded LDS row stride (floats) to avoid bank conflicts

// One workgroup = 16 chains (same class c), 4 waves; wave w owns N-tile j in [16w,16w+16).
// State: M[chain] scalar, p[chain][i] = exp(alpha - M) in LDS.
// Step: s = p x Ehat via V_WMMA_F32_16X16X4_F32;  alpha' = M + u[j] + log s[j],
// with u = colmax(Trans)+emis; renormalize with umax and smax (1 log/chain/step).
__global__ __launch_bounds__(128)
void crf_scan_wmma(const float* __restrict__ emis,
                   const float* __restrict__ trans,
                   const float* __restrict__ headt,
                   const float* __restrict__ tailt,
                   const long long* __restrict__ lens,
                   float* __restrict__ out)
{
  __shared__ float ldsP[16 * PSTR];     // p values, [chain][i]
  __shared__ float ldsU[16 * PSTR];     // u values, [chain][j]
  __shared__ float ldsTmax[Nn];         // column max of transitions
  __shared__ float ldsUmaxW[4 * 16];    // per-wave partial umax
  __shared__ float ldsSmaxW[4 * 16];    // per-wave partial smax
  __shared__ float ldsM[16];            // normalizers
  __shared__ float ldsFin[4 * 16];      // final partial sums

  const int tid = threadIdx.x;
  const int w   = tid >> 5;             // wave id 0..3  (owns columns 16w..16w+15)
  const int l   = tid & 31;
  const int n   = l & 15;
  const int g   = l >> 4;
  const int grp = blockIdx.x;           // 0..63
  const int c   = grp & 3;
  const int b0  = (grp >> 2) * 16;

  // ---------------- constant B fragments: Ehat[i][j] = exp(T[i][j] - colmax[j]) ----------------
  const int    j  = 16 * w + n;         // this lane's fixed column
  const float* tc = trans + (size_t)c * (Nn * Nn);
  float tmaxj = -__builtin_inff();
  #pragma unroll 8
  for (int i = 0; i < Nn; ++i) tmaxj = fmaxf(tmaxj, tc[i * Nn + j]);
  if (g == 0) ldsTmax[j] = tmaxj;

  v2f breg[16];                         // B tile k: K' = 2g + r, N = n  ->  i = 4k + 2g + r
  #pragma unroll
  for (int k = 0; k < 16; ++k) {
    const int i0 = 4 * k + 2 * g;
    breg[k].x = __expf(tc[(i0    ) * Nn + j] - tmaxj);
    breg[k].y = __expf(tc[(i0 + 1) * Nn + j] - tmaxj);
  }

  // lane roles
  const int ch1 = n;                    // phase-1 chain (one chain per lane, 2 lanes/chain)
  const int jb  = 16 * w + 8 * g;       // phase-1 column base (8 columns per lane)

  int lenr[8];                          // lengths for D-layout chains r + 8g
  #pragma unroll
  for (int r = 0; r < 8; ++r) lenr[r] = (int)lens[b0 + r + 8 * g];

  int lv = (int)lens[b0 + ch1];
  lv = max(lv, __shfl_xor(lv, 1, 32));
  lv = max(lv, __shfl_xor(lv, 2, 32));
  lv = max(lv, __shfl_xor(lv, 4, 32));
  lv = max(lv, __shfl_xor(lv, 8, 32));
  const int Lmax = lv;                  // uniform across block

  // ---------------- t = 0: alpha0 = emis[:,0] + head ----------------
  const float* erow = emis + (size_t)(b0 + ch1) * (Tn * Cn * Nn) + c * Nn + jb;
  float ev[8];
  {
    float4 q0 = *(const float4*)(erow);
    float4 q1 = *(const float4*)(erow + 4);
    ev[0]=q0.x; ev[1]=q0.y; ev[2]=q0.z; ev[3]=q0.w;
    ev[4]=q1.x; ev[5]=q1.y; ev[6]=q1.z; ev[7]=q1.w;
  }
  float a0[8];
  float lm = -__builtin_inff();
  #pragma unroll
  for (int m = 0; m < 8; ++m) {
    a0[m] = ev[m] + headt[c * Nn + jb + m];
    lm = fmaxf(lm, a0[m]);
  }
  lm = fmaxf(lm, __shfl_xor(lm, 16, 32));
  if (g == 0) ldsUmaxW[w * 16 + ch1] = lm;
  __syncthreads();
  const float M0 = fmaxf(fmaxf(ldsUmaxW[ch1], ldsUmaxW[16 + ch1]),
                         fmaxf(ldsUmaxW[32 + ch1], ldsUmaxW[48 + ch1]));
  #pragma unroll
  for (int m = 0; m < 8; ++m) ldsP[ch1 * PSTR + jb + m] = __expf(a0[m] - M0);
  if (w == 0 && g == 0) ldsM[ch1] = M0;
  __syncthreads();

  float Mreg[8];
  #pragma unroll
  for (int r = 0; r < 8; ++r) Mreg[r] = ldsM[r + 8 * g];
  float tmr[8];
  #pragma unroll
  for (int m = 0; m < 8; ++m) tmr[m] = ldsTmax[jb + m];

  // prefetch emissions for t = 1
  float4 ecA = *(const float4*)(erow + (Cn * Nn));
  float4 ecB = *(const float4*)(erow + (Cn * Nn) + 4);

  // ---------------- sequential scan ----------------
  for (int t = 1; t < Lmax; ++t) {
    // A fragments from LDS: A tile k: M = n, K' = 2g + r  ->  i = 4k + 2g + r
    v2f areg[16];
    #pragma unroll
    for (int k = 0; k < 16; ++k)
      areg[k] = *(const v2f*)&ldsP[n * PSTR + 4 * k + 2 * g];

    // prefetch t+1 (clamped, overlaps with WMMA/VALU below)
    const int    tn  = (t + 1 < Tn) ? (t + 1) : (Tn - 1);
    const float* epn = erow + (size_t)tn * (Cn * Nn);
    float4 enA = *(const float4*)(epn);
    float4 enB = *(const float4*)(epn + 4);

    // s[16 chains][16 cols of this wave] = p x Ehat  (fp32 WMMA chain over K)
    v8f acc = {0.f, 0.f, 0.f, 0.f, 0.f, 0.f, 0.f, 0.f};
    #pragma unroll
    for (int k = 0; k < 16; ++k)
      acc = __builtin_amdgcn_wmma_f32_16x16x4_f32(
              false, areg[k], false, breg[k], (short)0, acc, false, false);

    // phase 1: u = colmax + emis_t, per-chain partial max
    ev[0]=ecA.x; ev[1]=ecA.y; ev[2]=ecA.z; ev[3]=ecA.w;
    ev[4]=ecB.x; ev[5]=ecB.y; ev[6]=ecB.z; ev[7]=ecB.w;
    float uv[8];
    float lmax = -__builtin_inff();
    #pragma unroll
    for (int m = 0; m < 8; ++m) {
      uv[m] = ev[m] + tmr[m];
      lmax = fmaxf(lmax, uv[m]);
    }
    #pragma unroll
    for (int m = 0; m < 8; m += 2) {
      v2f tv = {uv[m], uv[m + 1]};
      *(v2f*)&ldsU[ch1 * PSTR + jb + m] = tv;
    }
    lmax = fmaxf(lmax, __shfl_xor(lmax, 16, 32));
    if (g == 0) ldsUmaxW[w * 16 + ch1] = lmax;
    __syncthreads();   // u + partial umax visible

    // phase 3a: combine umax, eu = exp(u - umax), partial smax (D layout: chain=r+8g, col=16w+n)
    float um[8], pnum[8];
    #pragma unroll
    for (int r = 0; r < 8; ++r) {
      const int ch = r + 8 * g;
      um[r] = fmaxf(fmaxf(ldsUmaxW[ch], ldsUmaxW[16 + ch]),
                    fmaxf(ldsUmaxW[32 + ch], ldsUmaxW[48 + ch]));
      const float sv = acc[r];
      const float eu = __expf(ldsU[ch * PSTR + 16 * w + n] - um[r]);
      pnum[r] = eu * sv;
      float pm = sv;
      pm = fmaxf(pm, __shfl_xor(pm, 1, 32));
      pm = fmaxf(pm, __shfl_xor(pm, 2, 32));
      pm = fmaxf(pm, __shfl_xor(pm, 4, 32));
      pm = fmaxf(pm, __shfl_xor(pm, 8, 32));
      if (n == 0) ldsSmaxW[w * 16 + ch] = pm;
    }
    __syncthreads();   // partial smax visible

    // phase 3b: finalize p' and M (length-masked)
    #pragma unroll
    for (int r = 0; r < 8; ++r) {
      const int ch = r + 8 * g;
      const float smax = fmaxf(fmaxf(ldsSmaxW[ch], ldsSmaxW[16 + ch]),
                               fmaxf(ldsSmaxW[32 + ch], ldsSmaxW[48 + ch]));
      const float pnew = pnum[r] * __builtin_amdgcn_rcpf(smax);
      const bool  act  = (t < lenr[r]);
      if (act) ldsP[ch * PSTR + 16 * w + n] = pnew;
      Mreg[r] = act ? (Mreg[r] + um[r] + __logf(smax)) : Mreg[r];
    }
    ecA = enA; ecB = enB;
    __syncthreads();   // p' ready for next iteration's A loads
  }

  // ---------------- finalize: out[b,c] = M + log(sum_j p[j] * exp(tail[j])) ----------------
  if (w == 0 && n == 0) {
    #pragma unroll
    for (int r = 0; r < 8; ++r) ldsM[r + 8 * g] = Mreg[r];
  }
  {
    float s = 0.f;
    #pragma unroll
    for (int m = 0; m < 8; ++m)
      s += ldsP[ch1 * PSTR + jb + m] * __expf(tailt[c * Nn + jb + m]);
    s += __shfl_xor(s, 16, 32);
    if (g == 0) ldsFin[w * 16 + ch1] = s;
  }
  __syncthreads();
  if (w == 0 && l < 16) {
    const float tot = ldsFin[l] + ldsFin[16 + l] + ldsFin[32 + l] + ldsFin[48 + l];
    out[(size_t)(b0 + l) * Cn + c] = ldsM[l] + __logf(tot);
  }
}

extern "C" void kernel_launch(void* const* d_in, const int* in_sizes, int n_in,
                              void* d_out, int out_size, void* d_ws, size_t ws_size,
                              hipStream_t stream) {
  (void)in_sizes; (void)n_in; (void)d_ws; (void)ws_size; (void)out_size;
  const float*     emis  = (const float*)d_in[0];
  const float*     trans = (const float*)d_in[1];
  const float*     headt = (const float*)d_in[2];
  const float*     tailt = (const float*)d_in[3];
  const long long* lens  = (const long long*)d_in[4];
  float* outp = (float*)d_out;

  dim3 grid(64);    // 16 b-groups x 4 classes
  dim3 block(128);  // 4 waves
  hipLaunchKernelGGL(crf_scan_wmma, grid, block, 0, stream,
                     emis, trans, headt, tailt, lens, outp);
}